// GCNLatencyModel_13589276524806
// MI455X (gfx1250) — compile-verified
//
#include <hip/hip_runtime.h>
#include <hip/hip_bf16.h>

typedef float v2f __attribute__((ext_vector_type(2)));
typedef float v8f __attribute__((ext_vector_type(8)));

#define FDIM 128

// ---------------------------------------------------------------------------
// h = x @ W1  via V_WMMA_F32_16X16X4_F32.
// One wave (32 lanes) computes one 16x16 tile of h. One block (8 waves)
// computes one 16-row strip (all 8 column tiles). K loop: 128/4 = 32 WMMAs.
//
// A (16x4 f32) lane layout (ISA 7.12.2): lane L holds
//   vgpr0 = A[L&15][kk + 2*(L>=16)], vgpr1 = A[L&15][kk + 2*(L>=16) + 1]
// B (4x16 f32): vgpr0 = B[kk + 2*(L>=16)][L&15], vgpr1 = next K row.
// C/D (16x16 f32): vgpr r -> row (r + 8*(L>=16)), col (L&15).
// ---------------------------------------------------------------------------
__global__ __launch_bounds__(256) void gcn_gemm_wmma(
    const float* __restrict__ x, const float* __restrict__ W,
    float* __restrict__ h, int n_nodes) {
  const int wave = threadIdx.x >> 5;     // 0..7 -> column tile
  const int lane = threadIdx.x & 31;
  const int tm = blockIdx.x;             // row tile
  if (tm * 16 + 16 > n_nodes) return;    // wave-uniform (N divisible by 16)

  const int row  = tm * 16 + (lane & 15);
  const int col  = wave * 16 + (lane & 15);
  const int khi  = (lane >> 4) << 1;     // 0 for lanes 0-15, 2 for lanes 16-31

  const float* xrow = x + (size_t)row * FDIM;

  v8f c = {};
#pragma unroll
  for (int kk = 0; kk < FDIM; kk += 4) {
    const int k = kk + khi;
    v2f a, b;
    a.x = xrow[k];
    a.y = xrow[k + 1];
    b.x = W[(size_t)k * FDIM + col];
    b.y = W[(size_t)(k + 1) * FDIM + col];
    c = __builtin_amdgcn_wmma_f32_16x16x4_f32(
        /*neg_a=*/false, a, /*neg_b=*/false, b,
        /*c_mod=*/(short)0, c, /*reuse_a=*/false, /*reuse_b=*/false);
  }

  const int mbase = tm * 16 + ((lane >> 4) << 3);
#pragma unroll
  for (int r = 0; r < 8; ++r) {
    h[(size_t)(mbase + r) * FDIM + col] = c[r];
  }
}

// ---------------------------------------------------------------------------
__global__ void zero_f(float* __restrict__ p, int n) {
  int i = blockIdx.x * blockDim.x + threadIdx.x;
  if (i < n) p[i] = 0.0f;
}

__global__ void deg_count(const int* __restrict__ dst, float* __restrict__ deg,
                          int n_edges) {
  int e = blockIdx.x * blockDim.x + threadIdx.x;
  if (e < n_edges) atomicAdd(&deg[dst[e]], 1.0f);
}

__global__ void compute_dinv(const float* __restrict__ deg,
                             float* __restrict__ dinv, int n) {
  int i = blockIdx.x * blockDim.x + threadIdx.x;
  if (i < n) dinv[i] = rsqrtf(deg[i] + 1.0f);
}

// agg = h * dinv^2 + b1   (self-loop + bias folded in; full overwrite)
__global__ void init_agg(const float* __restrict__ h,
                         const float* __restrict__ dinv,
                         const float* __restrict__ b1,
                         float* __restrict__ agg, int n_elems) {
  int i = blockIdx.x * blockDim.x + threadIdx.x;
  if (i < n_elems) {
    const int node = i >> 7;        // /128
    const int f    = i & (FDIM - 1);
    const float d  = dinv[node];
    agg[i] = h[i] * d * d + b1[f];
  }
}

// One wave per edge; each lane moves a float4 slice of the 128-wide message.
__global__ __launch_bounds__(256) void scatter_msgs(
    const int* __restrict__ src, const int* __restrict__ dst,
    const float* __restrict__ h, const float* __restrict__ dinv,
    float* __restrict__ agg, int n_edges) {
  const int wave = threadIdx.x >> 5;
  const int lane = threadIdx.x & 31;
  const int e = blockIdx.x * 8 + wave;
  if (e >= n_edges) return;
  const int s = src[e];
  const int d = dst[e];
  const float norm = dinv[s] * dinv[d];
  const float4 v = ((const float4*)(h + (size_t)s * FDIM))[lane];
  float* ad = agg + (size_t)d * FDIM + lane * 4;
  atomicAdd(ad + 0, v.x * norm);
  atomicAdd(ad + 1, v.y * norm);
  atomicAdd(ad + 2, v.z * norm);
  atomicAdd(ad + 3, v.w * norm);
}

// One wave per node: h2s[i] = dot(relu(agg[i,:]), W2); out[i] = h2s*dinv^2+b2.
__global__ __launch_bounds__(256) void node_relu_dot(
    const float* __restrict__ agg, const float* __restrict__ W2,
    const float* __restrict__ dinv, const float* __restrict__ b2,
    float* __restrict__ h2s, float* __restrict__ out, int n_nodes) {
  const int wave = threadIdx.x >> 5;
  const int lane = threadIdx.x & 31;
  const int i = blockIdx.x * 8 + wave;
  if (i >= n_nodes) return;
  const float4 a = ((const float4*)(agg + (size_t)i * FDIM))[lane];
  const float4 w = ((const float4*)W2)[lane];
  float sum = fmaxf(a.x, 0.0f) * w.x + fmaxf(a.y, 0.0f) * w.y +
              fmaxf(a.z, 0.0f) * w.z + fmaxf(a.w, 0.0f) * w.w;
#pragma unroll
  for (int off = 16; off > 0; off >>= 1) sum += __shfl_xor(sum, off, 32);
  if (lane == 0) {
    h2s[i] = sum;
    const float dv = dinv[i];
    out[i] = sum * dv * dv + b2[0];
  }
}

// Layer-2 message passing: scalar messages, one thread per edge.
__global__ void scatter_scalar(const int* __restrict__ src,
                               const int* __restrict__ dst,
                               const float* __restrict__ h2s,
                               const float* __restrict__ dinv,
                               float* __restrict__ out, int n_edges) {
  int e = blockIdx.x * blockDim.x + threadIdx.x;
  if (e < n_edges) {
    const int s = src[e];
    const int d = dst[e];
    atomicAdd(&out[d], h2s[s] * dinv[s] * dinv[d]);
  }
}

// ---------------------------------------------------------------------------
extern "C" void kernel_launch(void* const* d_in, const int* in_sizes, int n_in,
                              void* d_out, int out_size, void* d_ws,
                              size_t ws_size, hipStream_t stream) {
  const float* x   = (const float*)d_in[0];
  const int*   ei  = (const int*)d_in[1];
  const float* W1  = (const float*)d_in[2];
  const float* b1  = (const float*)d_in[3];
  const float* W2  = (const float*)d_in[4];
  const float* b2  = (const float*)d_in[5];
  float* out = (float*)d_out;

  const int n_nodes = in_sizes[0] / FDIM;   // 100000
  const int n_edges = in_sizes[1] / 2;      // 1600000
  const int* src = ei;
  const int* dst = ei + n_edges;

  // Workspace layout (floats): h[N*128] | agg[N*128] | deg[N] | dinv[N] | h2s[N]
  float* ws   = (float*)d_ws;
  float* h    = ws;
  float* agg  = h   + (size_t)n_nodes * FDIM;
  float* deg  = agg + (size_t)n_nodes * FDIM;
  float* dinv = deg + n_nodes;
  float* h2s  = dinv + n_nodes;

  // 1. deg = 0
  zero_f<<<(n_nodes + 255) / 256, 256, 0, stream>>>(deg, n_nodes);
  // 2. h = x @ W1  (WMMA f32)
  gcn_gemm_wmma<<<(n_nodes + 15) / 16, 256, 0, stream>>>(x, W1, h, n_nodes);
  // 3. deg counts over dst
  deg_count<<<(n_edges + 255) / 256, 256, 0, stream>>>(dst, deg, n_edges);
  // 4. dinv = rsqrt(deg + 1)
  compute_dinv<<<(n_nodes + 255) / 256, 256, 0, stream>>>(deg, dinv, n_nodes);
  // 5. agg = h*dinv^2 + b1
  {
    const int n_elems = n_nodes * FDIM;
    init_agg<<<(n_elems + 255) / 256, 256, 0, stream>>>(h, dinv, b1, agg, n_elems);
  }
  // 6. agg += scatter(h[src] * norm)
  scatter_msgs<<<(n_edges + 7) / 8, 256, 0, stream>>>(src, dst, h, dinv, agg,
                                                      n_edges);
  // 7. h2s = relu(agg) . W2 ; out = h2s*dinv^2 + b2
  node_relu_dot<<<(n_nodes + 7) / 8, 256, 0, stream>>>(agg, W2, dinv, b2, h2s,
                                                       out, n_nodes);
  // 8. out += scatter(h2s[src] * norm)
  scatter_scalar<<<(n_edges + 255) / 256, 256, 0, stream>>>(src, dst, h2s, dinv,
                                                            out, n_edges);
}